// AttnMatching_13580686590581
// MI455X (gfx1250) — compile-verified
//
#include <hip/hip_runtime.h>
#include <hip/hip_bf16.h>

typedef __attribute__((ext_vector_type(2))) float v2f;
typedef __attribute__((ext_vector_type(4))) float v4f;
typedef __attribute__((ext_vector_type(8))) float v8f;

#define LOC_MAX 50000
#define EMB_DIM 128
#define MAX_LEN 100
#define BATCH   16

#define WAVES_PER_BLOCK 4
#define ROW_STRIDE 132   // 128 + 4 floats pad -> bank-conflict-free strided reads

// ---------------------------------------------------------------------------
// Kernel 1: V[n][d] = sum_t value_w[t] * self_attn[n][t][d]   (16 x 128)
// ---------------------------------------------------------------------------
__global__ void __launch_bounds__(EMB_DIM)
fold_value_kernel(const float* __restrict__ self_attn,   // [16][100][128]
                  const float* __restrict__ value_w,     // [100]
                  float* __restrict__ V)                 // [16][128]
{
    const int n = blockIdx.x;
    const int d = threadIdx.x;
    const float* sa = self_attn + (size_t)n * MAX_LEN * EMB_DIM + d;
    float acc = 0.0f;
    #pragma unroll 4
    for (int t = 0; t < MAX_LEN; ++t)
        acc = fmaf(value_w[t], sa[(size_t)t * EMB_DIM], acc);
    V[n * EMB_DIM + d] = acc;
}

// ---------------------------------------------------------------------------
// Kernel 2: out[n][l] = sum_d V[n][d] * emb_table[1+l][d]
// One wave32 -> one 16(batch) x 16(loc) f32 tile via V_WMMA_F32_16X16X4_F32,
// K = 128 swept in 32 steps. emb tile staged through padded LDS with
// coalesced b128 loads.
// ---------------------------------------------------------------------------
__global__ void __launch_bounds__(WAVES_PER_BLOCK * 32)
attn_gemm_kernel(const float* __restrict__ V,          // [16][128]
                 const float* __restrict__ emb_table,  // [50001][128], rows 1..50000 used
                 float* __restrict__ out)              // [16][50000]
{
    __shared__ float sm[WAVES_PER_BLOCK * 16 * ROW_STRIDE];

    const int tid    = threadIdx.x;
    const int wave   = tid >> 5;
    const int lane   = tid & 31;
    const int lane16 = lane & 15;
    const int half   = lane >> 4;

    const int tile   = blockIdx.x * WAVES_PER_BLOCK + wave;
    const int l_base = tile * 16;                  // 50000 % 16 == 0 -> clean tiles
    const bool valid = (l_base < LOC_MAX);

    float* smw = sm + wave * 16 * ROW_STRIDE;

    if (valid) {
        // Cooperative coalesced load: 16 rows x 128 f32 = 512 float4 chunks.
        // Memory chunk idx -> row = idx/32, col = (idx%32)*4.
        const float* src = emb_table + (size_t)(l_base + 1) * EMB_DIM;
        #pragma unroll
        for (int i = 0; i < 16; ++i) {
            const int idx = i * 32 + lane;
            const int r   = idx >> 5;
            const int c   = (idx & 31) * 4;
            v4f val = *(const v4f*)(src + (size_t)idx * 4);
            *(v4f*)(smw + r * ROW_STRIDE + c) = val;
        }
    }
    __syncthreads();
    if (!valid) return;

    // WMMA f32 16x16x4 operand layout (wave32):
    //   A (16x4):  lane holds A[M=lane16][K = 2*half + v],  v in {0,1}
    //   B (4x16):  lane holds B[K = 2*half + v][N=lane16]
    //   C/D:       VGPR v holds D[M = v + 8*half][N = lane16]
    v8f acc = {};
    const float* aV = V   + lane16 * EMB_DIM    + 2 * half;  // A = V (M=batch)
    const float* bS = smw + lane16 * ROW_STRIDE + 2 * half;  // B = emb tile (N=loc)

    #pragma unroll
    for (int kb = 0; kb < EMB_DIM; kb += 4) {
        v2f a = *(const v2f*)(aV + kb);
        v2f b = *(const v2f*)(bS + kb);
        acc = __builtin_amdgcn_wmma_f32_16x16x4_f32(
            /*neg_a=*/false, a, /*neg_b=*/false, b,
            /*c_mod=*/(short)0, acc, /*reuse_a=*/false, /*reuse_b=*/false);
    }

    // Scatter D: row M = v + 8*half (batch), col N = l_base + lane16.
    float* o = out + (size_t)(8 * half) * LOC_MAX + (l_base + lane16);
    #pragma unroll
    for (int v = 0; v < 8; ++v)
        o[(size_t)v * LOC_MAX] = acc[v];
}

// ---------------------------------------------------------------------------
// Inputs (setup_inputs order):
//   d_in[0] self_attn  f32 [16][100][128]
//   d_in[1] mat2       f32 (unused)
//   d_in[2] traj       int (unused)
//   d_in[3] emb_table  f32 [50001][128]
//   d_in[4] value_w    f32 [1][100]
// Output: f32 [16][50000]
// ---------------------------------------------------------------------------
extern "C" void kernel_launch(void* const* d_in, const int* in_sizes, int n_in,
                              void* d_out, int out_size, void* d_ws, size_t ws_size,
                              hipStream_t stream) {
    const float* self_attn = (const float*)d_in[0];
    const float* emb_table = (const float*)d_in[3];
    const float* value_w   = (const float*)d_in[4];
    float*       out       = (float*)d_out;
    float*       V         = (float*)d_ws;   // 16*128 f32 = 8 KB scratch

    fold_value_kernel<<<BATCH, EMB_DIM, 0, stream>>>(self_attn, value_w, V);

    const int n_tiles = LOC_MAX / 16;                                  // 3125
    const int n_blocks = (n_tiles + WAVES_PER_BLOCK - 1) / WAVES_PER_BLOCK; // 782
    attn_gemm_kernel<<<n_blocks, WAVES_PER_BLOCK * 32, 0, stream>>>(V, emb_table, out);
}